// Sparse_Three_Sum_86973087744647
// MI455X (gfx1250) — compile-verified
//
#include <hip/hip_runtime.h>
#include <math.h>

typedef __attribute__((ext_vector_type(2))) float v2f;
typedef __attribute__((ext_vector_type(8))) float v8f;

#define N_NODES 50000
#define F_IN    128
#define HDIM    128
#define CDIM    40
#define E_EDGES 600000

// out[M,Nd] = X[M,K] @ W[K,Nd] + (b0 + b1 + b2)   (bias pointers optional)
// grid.x = M/16 row tiles; blockDim.x = 32 * ceil(Nd/16); one wave per 16x16 tile.
// A tile staged in LDS (conflict-free padded stride); V_WMMA_F32_16X16X4_F32,
// K fully unrolled -> constant-offset loads, no divergent branches in the loop.
template<int K, int Nd>
__global__ __launch_bounds__(256) void gemm_wmma_f32(
    const float* __restrict__ X, const float* __restrict__ W,
    const float* __restrict__ b0, const float* __restrict__ b1, const float* __restrict__ b2,
    float* __restrict__ out)
{
  constexpr int CT  = (Nd + 15) / 16;   // col tiles == waves per block
  constexpr int LDA = K + 4;            // padded LDS stride (dwords): banks 4l+k+{0..3} -> conflict-free
  __shared__ float As[16 * LDA];

  const int tid  = threadIdx.x;
  const int lane = tid & 31;
  const int wave = tid >> 5;
  const int m0   = blockIdx.x << 4;

  // Cooperative load of the 16xK X tile into LDS (float4 chunks).
  for (int i = tid; i < (16 * K) / 4; i += 32 * CT) {
    const int r = (i * 4) / K;
    const int c = (i * 4) % K;
    *(float4*)(&As[r * LDA + c]) = *(const float4*)(X + (size_t)(m0 + r) * K + c);
  }
  __syncthreads();

  const int hf = lane >> 4;             // 0: K pair {0,1}, 1: K pair {2,3}
  const int l  = lane & 15;
  const int n0 = wave << 4;
  const bool colOK = (n0 + l) < Nd;
  const int col = colOK ? (n0 + l) : (Nd - 1);   // clamp: garbage cols never stored

  const float* ap = &As[l * LDA + (hf << 1)];            // A: ds_load_b64 per step
  const float* wp = W + (size_t)(hf << 1) * Nd + col;    // B: two b32 loads per step

  v8f acc = {};
  #pragma unroll
  for (int k = 0; k < K; k += 4) {
    v2f a = *(const v2f*)(ap + k);
    v2f b;
    b.x = wp[k * Nd];            // row k + 2*hf      (constant offsets)
    b.y = wp[k * Nd + Nd];       // row k + 2*hf + 1
    acc = __builtin_amdgcn_wmma_f32_16x16x4_f32(false, a, false, b,
                                                (short)0, acc, false, false);
  }

  if (colOK) {
    float bias = 0.0f;
    if (b0) bias += b0[col];
    if (b1) bias += b1[col];
    if (b2) bias += b2[col];
    // C/D layout: VGPR r -> row m0 + r + 8*hf, col n0 + l
    float* o = out + (size_t)(m0 + (hf << 3)) * Nd + col;
    #pragma unroll
    for (int r = 0; r < 8; ++r)
      o[(size_t)r * Nd] = acc[r] + bias;
  }
}

// out[dst[e], :] += ew[e] * h[src[e], :]
// One thread: 4 consecutive features (float4 gather + 4 global_atomic_add_f32).
// For F=128 one wave covers one edge row -> coalesced 512B gather.
template<int F>
__global__ void scatter_add(const float* __restrict__ h, const int* __restrict__ ei,
                            const float* __restrict__ ew, float* __restrict__ out,
                            int nE)
{
  constexpr int per = F / 4;
  const int tid = blockIdx.x * blockDim.x + threadIdx.x;
  if (tid >= nE * per) return;
  const int e = tid / per;
  const int c = (tid - e * per) << 2;
  const int src = ei[e];                 // edge_index[0, e]
  const int dst = ei[nE + e];            // edge_index[1, e]
  const float w = ew[e];
  const float4 v = *(const float4*)(h + (size_t)src * F + c);
  float* o = out + (size_t)dst * F + c;
  atomicAdd(o + 0, w * v.x);
  atomicAdd(o + 1, w * v.y);
  atomicAdd(o + 2, w * v.z);
  atomicAdd(o + 3, w * v.w);
}

// In-place per-row log_softmax, one thread per row (C=40).
__global__ void log_softmax_rows(float* __restrict__ out, int M, int C)
{
  const int r = blockIdx.x * blockDim.x + threadIdx.x;
  if (r >= M) return;
  float* p = out + (size_t)r * C;
  float mx = -INFINITY;
  for (int c = 0; c < C; ++c) mx = fmaxf(mx, p[c]);
  float s = 0.0f;
  for (int c = 0; c < C; ++c) s += expf(p[c] - mx);
  const float lse = mx + logf(s);
  for (int c = 0; c < C; ++c) p[c] -= lse;
}

template<int K, int Nd>
static void run_block(const float* Xin, float* O, float* t1, float* t2,
                      const int* ei1, const float* ew1,
                      const int* ei2, const float* ew2,
                      const float* Wln, const float* bln,
                      const float* Wc1, const float* bc1,
                      const float* Wc2, const float* bc2,
                      hipStream_t stream)
{
  constexpr int CT = (Nd + 15) / 16;
  const dim3 bs(32 * CT);
  const int rowTiles = N_NODES / 16;   // 3125 (exact)
  gemm_wmma_f32<K, Nd><<<rowTiles, bs, 0, stream>>>(Xin, Wc1, nullptr, nullptr, nullptr, t1);
  gemm_wmma_f32<K, Nd><<<rowTiles, bs, 0, stream>>>(Xin, Wc2, nullptr, nullptr, nullptr, t2);
  // dense branch + all three biases folded in; fully initializes O before scatters
  gemm_wmma_f32<K, Nd><<<rowTiles, bs, 0, stream>>>(Xin, Wln, bln, bc1, bc2, O);
  constexpr int grid = (int)(((long)E_EDGES * (Nd / 4) + 255) / 256);
  scatter_add<Nd><<<grid, 256, 0, stream>>>(t1, ei1, ew1, O, E_EDGES);
  scatter_add<Nd><<<grid, 256, 0, stream>>>(t2, ei2, ew2, O, E_EDGES);
}

extern "C" void kernel_launch(void* const* d_in, const int* in_sizes, int n_in,
                              void* d_out, int out_size, void* d_ws, size_t ws_size,
                              hipStream_t stream)
{
  const float* x   = (const float*)d_in[0];
  const int*   ei1 = (const int*)  d_in[1];
  const float* ew1 = (const float*)d_in[2];
  const int*   ei2 = (const int*)  d_in[3];
  const float* ew2 = (const float*)d_in[4];

  // Params in dict order: (W_ln, b_ln, W_c1, b_c1, W_c2, b_c2) x 3 layers
  const float* Wp[9]; const float* bp[9];
  for (int i = 0; i < 9; ++i) {
    Wp[i] = (const float*)d_in[5 + 2 * i];
    bp[i] = (const float*)d_in[6 + 2 * i];
  }

  float* bufA = (float*)d_ws;                         // block1 output [N,128]
  float* bufB = bufA + (size_t)N_NODES * HDIM;        // block2 output [N,128]
  float* t1   = bufB + (size_t)N_NODES * HDIM;        // conv1 pre-scatter
  float* t2   = t1   + (size_t)N_NODES * HDIM;        // conv2 pre-scatter

  run_block<F_IN, HDIM>(x,    bufA,          t1, t2, ei1, ew1, ei2, ew2,
                        Wp[0], bp[0], Wp[1], bp[1], Wp[2], bp[2], stream);
  run_block<HDIM, HDIM>(bufA, bufB,          t1, t2, ei1, ew1, ei2, ew2,
                        Wp[3], bp[3], Wp[4], bp[4], Wp[5], bp[5], stream);
  run_block<HDIM, CDIM>(bufB, (float*)d_out, t1, t2, ei1, ew1, ei2, ew2,
                        Wp[6], bp[6], Wp[7], bp[7], Wp[8], bp[8], stream);

  log_softmax_rows<<<(N_NODES + 255) / 256, 256, 0, stream>>>((float*)d_out, N_NODES, CDIM);
}